// YoloLossLayer_1417339207726
// MI455X (gfx1250) — compile-verified
//
#include <hip/hip_runtime.h>
#include <math.h>

// ---------------- geometry constants (from reference) ----------------
#define GWD   52
#define HW    2704          // 52*52
#define NA    5
#define NC    80
#define NB    32
#define NCHAN 85            // 5 + NC
#define CELLS_PER_BLOCK 256
#define NCHUNK_HW 11        // ceil(2704/256)
#define NBLOCKS (NB * NA * NCHUNK_HW)   // 1760

#define AS1 __attribute__((address_space(1)))
#define AS3 __attribute__((address_space(3)))

// The async-to-LDS builtin takes pointers to GCC-style int vectors (16B).
typedef int v4i __attribute__((vector_size(16)));

// ---------------- CDNA5 async global->LDS copy (b128) ----------------
__device__ __forceinline__ void async_cp16(const float* g, float* l) {
#if __has_builtin(__builtin_amdgcn_global_load_async_to_lds_b128)
  __builtin_amdgcn_global_load_async_to_lds_b128(
      (AS1 v4i*)(AS1 void*)g, (AS3 v4i*)(AS3 void*)l,
      /*offset=*/0, /*cpol=*/0);
#else
  unsigned ldsOff = (unsigned)(unsigned long long)(AS3 float*)l;
  asm volatile("global_load_async_to_lds_b128 %0, %1, off"
               :: "v"(ldsOff), "v"(g) : "memory");
#endif
}

template <int N>
__device__ __forceinline__ void wait_async() {
#if __has_builtin(__builtin_amdgcn_s_wait_asynccnt)
  __builtin_amdgcn_s_wait_asynccnt(N);
#else
  asm volatile("s_wait_asynccnt %0" :: "n"(N) : "memory");
#endif
}

__device__ __forceinline__ float fast_sig(float x) {
  return __builtin_amdgcn_rcpf(1.0f + __expf(-x));
}

// Process one 16-channel chunk of staged logits: online logsumexp merge.
// Chunk C is ready once the first 4*(C+1) async ops per wave complete
// (in-order completion) => allowed outstanding = 16 - 4*C.
template <int C>
__device__ __forceinline__ void do_chunk(const float* __restrict__ smem, int tid,
                                         int tcls, float& m, float& s,
                                         float& logit_t) {
  wait_async<16 - 4 * C>();
  __syncthreads();                       // all waves' chunk-C tiles visible
  float mc = -3.4e38f;
#pragma unroll
  for (int k = 0; k < 16; ++k)
    mc = fmaxf(mc, smem[(C * 16 + k) * 256 + tid]);
  float sc = 0.0f;
#pragma unroll
  for (int k = 0; k < 16; ++k) {
    float v = smem[(C * 16 + k) * 256 + tid];
    sc += __expf(v - mc);
    if (C * 16 + k == tcls) logit_t = v;
  }
  float M = fmaxf(m, mc);
  s = s * __expf(m - M) + sc * __expf(mc - M);
  m = M;
}

// ---------------- kernel 1: per-block partial loss ----------------
__global__ __launch_bounds__(256)
void yolo_loss_partial(const float* __restrict__ net,   // (B,A,85,HW) flat
                       const float* __restrict__ tgt,   // (B,A,6,HW)
                       const float* __restrict__ amask, // (B,A,HW)
                       const float* __restrict__ omask, // (B,A,HW)
                       const float* __restrict__ anc,   // (5,2)
                       float* __restrict__ ws) {
  extern __shared__ float smem[];        // 80 channels * 256 cells = 80 KB
  const int tid   = threadIdx.x;
  const int bx    = blockIdx.x;
  const int chunk = bx % NCHUNK_HW;
  const int ba    = bx / NCHUNK_HW;      // b*5 + a
  const int a     = ba % NA;
  const int hw0   = chunk * CELLS_PER_BLOCK;
  const int hw    = hw0 + tid;
  const bool valid = (hw < HW);
  const int hwc   = valid ? hw : (HW - 1);

  const float* gbase = net + (size_t)ba * (NCHAN * HW);

  // ---- stage 80 logit channels into LDS via async DMA (20 x b128/thread) ---
  // seg r*256+tid -> channel seg/64, 4-float column group (seg&63)*4.
  // Wave w's r-th op covers channel 4r + w/2, so the first 4(c+1) ops of every
  // wave cover channels [0, 16(c+1)) -> threshold waits in do_chunk<>.
#pragma unroll
  for (int r = 0; r < 20; ++r) {
    int seg  = r * 256 + tid;
    int ch   = seg >> 6;                 // 0..79
    int off  = (seg & 63) << 2;          // 0..252
    int offc = min(off, HW - 4 - hw0);   // clamp tail block inside channel row
    const float* g = gbase + (size_t)(5 + ch) * HW + hw0 + offc;
    float* l = smem + ch * 256 + off;
    async_cp16(g, l);
  }

  // ---- direct coalesced loads overlap with the DMA ----
  const float p0 = gbase[0 * HW + hwc];
  const float p1 = gbase[1 * HW + hwc];
  const float p2 = gbase[2 * HW + hwc];
  const float p3 = gbase[3 * HW + hwc];
  const float p4 = gbase[4 * HW + hwc];
  const size_t tb = (size_t)ba * 6 * HW;
  const float t0 = tgt[tb + 0 * HW + hwc];
  const float t1 = tgt[tb + 1 * HW + hwc];
  const float t2 = tgt[tb + 2 * HW + hwc];
  const float t3 = tgt[tb + 3 * HW + hwc];
  const float t4 = tgt[tb + 4 * HW + hwc];
  const float t5 = tgt[tb + 5 * HW + hwc];
  const float am = amask[(size_t)ba * HW + hwc];
  const float om = omask[(size_t)ba * HW + hwc];
  const float aw = anc[2 * a + 0];
  const float ah = anc[2 * a + 1];

  // ---- box / conf terms (NET/gw = 416/52 = 8) ----
  const float x    = fast_sig(p0);
  const float y    = fast_sig(p1);
  const float conf = fast_sig(p4);
  const float ew  = __expf(p2) * aw;     // pred box w (pixels)
  const float eh  = __expf(p3) * ah;
  const float gw_ = __expf(t2) * aw;     // gt box w
  const float gh_ = __expf(t3) * ah;
  const float inv416 = 1.0f / 416.0f;

  const float dx = x - t0, dy = y - t1;
  const float xy_l = (dx * dx + dy * dy) * am;
  const float dws = (ew - gw_) * inv416;
  const float dhs = (eh - gh_) * inv416;
  const float wh_l = (dws * dws + dhs * dhs) * am;
  const float coords = xy_l + wh_l;

  const float dc = conf - t4;
  const float d2 = dc * dc;
  const float obj = d2 * am;

  const int row = hwc / GWD;
  const int col = hwc - row * GWD;
  const float cxp = (x + (float)col) * 8.0f, cyp = (y + (float)row) * 8.0f;
  const float px1 = cxp - 0.5f * ew, py1 = cyp - 0.5f * eh;
  const float px2 = px1 + ew,        py2 = py1 + eh;
  const float cxg = (t0 + (float)col) * 8.0f, cyg = (t1 + (float)row) * 8.0f;
  const float gx1 = cxg - 0.5f * gw_, gy1 = cyg - 0.5f * gh_;
  const float gx2 = gx1 + gw_,        gy2 = gy1 + gh_;
  const float iw = fmaxf(fminf(px2, gx2) - fmaxf(px1, gx1), 0.0f);
  const float ih = fmaxf(fminf(py2, gy2) - fmaxf(py1, gy1), 0.0f);
  const float inter = iw * ih;
  const float uni = ew * eh + gw_ * gh_ - inter + 1e-9f;
  const float iou = inter * __builtin_amdgcn_rcpf(uni);
  const float lt = (iou < 0.5f) ? 1.0f : 0.0f;
  const float noobj = d2 * (1.0f - om) + d2 * (1.0f - am) * om * lt;

  // ---- class CE from staged logits (online logsumexp over 5 chunks) ----
  const int tcls = (int)t5;
  float m = -3.4e38f, s = 0.0f, logit_t = 0.0f;
  do_chunk<0>(smem, tid, tcls, m, s, logit_t);
  do_chunk<1>(smem, tid, tcls, m, s, logit_t);
  do_chunk<2>(smem, tid, tcls, m, s, logit_t);
  do_chunk<3>(smem, tid, tcls, m, s, logit_t);
  do_chunk<4>(smem, tid, tcls, m, s, logit_t);
  const float ce = (m + __logf(s)) - logit_t;
  const float clas = ce * am;

  float total = clas + 5.0f * coords + 5.0f * obj + noobj;
  if (!valid) total = 0.0f;

  // ---- deterministic block reduction (reuse consumed chunk-0 LDS) ----
  smem[tid] = total;
  __syncthreads();
#pragma unroll
  for (int o = 128; o > 0; o >>= 1) {
    if (tid < o) smem[tid] += smem[tid + o];
    __syncthreads();
  }
  if (tid == 0) ws[bx] = smem[0];
}

// ---------------- kernel 2: fixed-order final reduction ----------------
__global__ __launch_bounds__(256)
void yolo_loss_reduce(const float* __restrict__ ws, float* __restrict__ out,
                      int n) {
  __shared__ float red[256];
  const int tid = threadIdx.x;
  float s = 0.0f;
  for (int i = tid; i < n; i += 256) s += ws[i];
  red[tid] = s;
  __syncthreads();
#pragma unroll
  for (int o = 128; o > 0; o >>= 1) {
    if (tid < o) red[tid] += red[tid + o];
    __syncthreads();
  }
  if (tid == 0) out[0] = red[0];
}

extern "C" void kernel_launch(void* const* d_in, const int* in_sizes, int n_in,
                              void* d_out, int out_size, void* d_ws,
                              size_t ws_size, hipStream_t stream) {
  (void)in_sizes; (void)n_in; (void)out_size; (void)ws_size;
  const float* net = (const float*)d_in[0];
  const float* tgt = (const float*)d_in[1];
  const float* am  = (const float*)d_in[2];
  const float* om  = (const float*)d_in[3];
  const float* anc = (const float*)d_in[4];
  float* out = (float*)d_out;
  float* ws  = (float*)d_ws;

  const size_t shmem = (size_t)NC * CELLS_PER_BLOCK * sizeof(float);  // 80 KB
  yolo_loss_partial<<<NBLOCKS, CELLS_PER_BLOCK, shmem, stream>>>(
      net, tgt, am, om, anc, ws);
  yolo_loss_reduce<<<1, 256, 0, stream>>>(ws, out, NBLOCKS);
}